// L3_86990267613961
// MI455X (gfx1250) — compile-verified
//
#include <hip/hip_runtime.h>

typedef float v2f __attribute__((ext_vector_type(2)));
typedef float v8f __attribute__((ext_vector_type(8)));

#define PITCH 132  // even (keeps b64 frags 8B-aligned); 132%64=4 -> b64 frag
                   // loads across a wave cover all 64 LDS banks exactly once

// Fused Kronecker-factorized linear:
//   out[b, g2, g1] = sum_{r,f} x[b, r, f] * w0[f, g1] * w1[r, g2]
// Per batch element: T = X_b @ W0 ; Out_b = W1^T @ T.
// Weights and T are kept TRANSPOSED in LDS so every WMMA fragment is a
// contiguous K-pair -> single ds_load_b64 straight into an even VGPR pair.
__global__ __launch_bounds__(256) void faclin_wmma_f32_kernel(
    const float* __restrict__ x, const float* __restrict__ w0,
    const float* __restrict__ w1, float* __restrict__ out) {
  extern __shared__ float smem[];
  float* sX   = smem;                 // X row-major (later T^T), PITCH*128
  float* sW0t = smem + PITCH * 128;   // W0^T: sW0t[g1*PITCH + f]
  float* sW1t = sW0t + PITCH * 128;   // W1^T: sW1t[g2*PITCH + r]

  const int tid  = threadIdx.x;
  const int lane = tid & 31;
  const int wave = tid >> 5;
  const int m    = lane & 15;  // fragment row/col index
  const int h    = lane >> 4;  // half-wave select
  const int b    = blockIdx.x;

  const float* xb = x + (size_t)b * 16384;

  // ---- staging: X row-major (coalesced b128 both sides) -------------------
  for (int i = tid; i < 4096; i += 256) {
    const float4 v = ((const float4*)xb)[i];
    const int r = i >> 5;
    const int f = (i & 31) << 2;
    *(float4*)(sX + r * PITCH + f) = v;
  }
  // ---- staging: weights transposed (coalesced reads, scattered LDS b32) --
  for (int i = tid; i < 4096; i += 256) {
    const int f = i >> 5;
    const int g = (i & 31) << 2;
    const float4 v0 = ((const float4*)w0)[i];
    sW0t[(g + 0) * PITCH + f] = v0.x;
    sW0t[(g + 1) * PITCH + f] = v0.y;
    sW0t[(g + 2) * PITCH + f] = v0.z;
    sW0t[(g + 3) * PITCH + f] = v0.w;
    const float4 v1 = ((const float4*)w1)[i];
    sW1t[(g + 0) * PITCH + f] = v1.x;
    sW1t[(g + 1) * PITCH + f] = v1.y;
    sW1t[(g + 2) * PITCH + f] = v1.z;
    sW1t[(g + 3) * PITCH + f] = v1.w;
  }
  __syncthreads();

  const int m0 = wave * 16;  // this wave owns output rows [m0, m0+16)

  // ---- GEMM 1: T[r, g1] = sum_f X[r, f] * W0[f, g1] ----------------------
  v8f acc[8];
#pragma unroll
  for (int t = 0; t < 8; ++t)
#pragma unroll
    for (int e = 0; e < 8; ++e) acc[t][e] = 0.0f;

  for (int kk = 0; kk < 128; kk += 4) {
    const int ka = kk + 2 * h;  // this half-wave's K pair
    const v2f a = *(const v2f*)(sX + (m0 + m) * PITCH + ka);   // X[m0+m, ka..+1]
#pragma unroll
    for (int ni = 0; ni < 8; ++ni) {
      const v2f bf = *(const v2f*)(sW0t + (ni * 16 + m) * PITCH + ka);
      acc[ni] = __builtin_amdgcn_wmma_f32_16x16x4_f32(
          false, a, false, bf, (short)0, acc[ni], false, false);
    }
  }

  // ---- write T^T over sX: lane writes 8 consecutive floats (2x b128) -----
  __syncthreads();
#pragma unroll
  for (int ni = 0; ni < 8; ++ni) {
    float* dst = sX + (ni * 16 + m) * PITCH + m0 + 8 * h;  // T^T[g1, r-base]
    float4 lo, hi;
    lo.x = acc[ni][0]; lo.y = acc[ni][1]; lo.z = acc[ni][2]; lo.w = acc[ni][3];
    hi.x = acc[ni][4]; hi.y = acc[ni][5]; hi.z = acc[ni][6]; hi.w = acc[ni][7];
    *(float4*)dst = lo;
    *(float4*)(dst + 4) = hi;
  }
  __syncthreads();

  // ---- GEMM 2: Out[g2, g1] = sum_r W1[r, g2] * T[r, g1] ------------------
#pragma unroll
  for (int t = 0; t < 8; ++t)
#pragma unroll
    for (int e = 0; e < 8; ++e) acc[t][e] = 0.0f;

  for (int kk = 0; kk < 128; kk += 4) {
    const int ka = kk + 2 * h;
    const v2f a = *(const v2f*)(sW1t + (m0 + m) * PITCH + ka);  // W1^T[m0+m, ka..+1]
#pragma unroll
    for (int ni = 0; ni < 8; ++ni) {
      const v2f bf = *(const v2f*)(sX + (ni * 16 + m) * PITCH + ka);  // T^T[g1, ka..+1]
      acc[ni] = __builtin_amdgcn_wmma_f32_16x16x4_f32(
          false, a, false, bf, (short)0, acc[ni], false, false);
    }
  }

  // ---- store Out_b, coalesced along g1 ------------------------------------
  float* ob = out + (size_t)b * 16384;
#pragma unroll
  for (int ni = 0; ni < 8; ++ni)
#pragma unroll
    for (int v = 0; v < 8; ++v)
      ob[(m0 + v + 8 * h) * 128 + ni * 16 + m] = acc[ni][v];
}

extern "C" void kernel_launch(void* const* d_in, const int* in_sizes, int n_in,
                              void* d_out, int out_size, void* d_ws,
                              size_t ws_size, hipStream_t stream) {
  (void)in_sizes; (void)n_in; (void)out_size; (void)d_ws; (void)ws_size;
  const float* x  = (const float*)d_in[0];
  const float* w0 = (const float*)d_in[1];
  const float* w1 = (const float*)d_in[2];
  float* out = (float*)d_out;

  const size_t shmem = (size_t)(3 * PITCH * 128) * sizeof(float);  // ~198 KB
  // Opt in to >64KB dynamic LDS (320 KB available per WGP on gfx1250).
  (void)hipFuncSetAttribute((const void*)faclin_wmma_f32_kernel,
                            hipFuncAttributeMaxDynamicSharedMemorySize,
                            (int)shmem);
  faclin_wmma_f32_kernel<<<4096, 256, shmem, stream>>>(x, w0, w1, out);
}